// GAT_22857815949369
// MI455X (gfx1250) — compile-verified
//
#include <hip/hip_runtime.h>
#include <math.h>

typedef __attribute__((ext_vector_type(2))) float v2f;
typedef __attribute__((ext_vector_type(8))) float v8f;

#define NEG_SLOPE 0.2f
#define PYG_EPS   1e-16f

// ---------------------------------------------------------------------------
// GEMM: Hout[N x M] = X[N x K] * W[K x M], exact fp32 via V_WMMA_F32_16X16X4_F32.
// One wave32 computes one 16x16 output tile. N % 16 == 0, M % 16 == 0, K % 4 == 0.
// A fragment: lanes 0-15 hold (M=lane, K=k0+0..1), lanes 16-31 hold (M=lane-16,
// K=k0+2..3) -> contiguous float2 load from row-major X.
// B fragment: VGPR0 = row k0+kb, VGPR1 = row k0+kb+1, striped over lane&15.
// ---------------------------------------------------------------------------
__global__ __launch_bounds__(256) void gemm_wmma_f32(
    const float* __restrict__ X, const float* __restrict__ W,
    float* __restrict__ Hout, int Nrows, int K, int M)
{
  const int tilesN = M >> 4;
  const int tilesTotal = (Nrows >> 4) * tilesN;
  const int wave = blockIdx.x * (blockDim.x >> 5) + (threadIdx.x >> 5);
  if (wave >= tilesTotal) return;          // wave-uniform: EXEC stays all-ones
  const int tm = wave / tilesN;
  const int tn = wave - tm * tilesN;
  const int lane = threadIdx.x & 31;
  const int half = lane >> 4;              // 0: K pair {0,1}, 1: K pair {2,3}
  const int l15  = lane & 15;
  const int kb   = half << 1;

  const float* xp = X + (size_t)((tm << 4) + l15) * K + kb;   // 8B aligned
  const float* wp = W + (size_t)kb * M + ((tn << 4) + l15);

  v8f acc = {0.f, 0.f, 0.f, 0.f, 0.f, 0.f, 0.f, 0.f};
  for (int k0 = 0; k0 < K; k0 += 4) {
    v2f a = *(const v2f*)xp;               // X[row][k0+kb], X[row][k0+kb+1]
    v2f b;
    b.x = wp[0];                           // W[k0+kb  ][col]
    b.y = wp[M];                           // W[k0+kb+1][col]
    acc = __builtin_amdgcn_wmma_f32_16x16x4_f32(
        /*neg_a=*/false, a, /*neg_b=*/false, b,
        /*c_mod=*/(short)0, acc, /*reuse_a=*/false, /*reuse_b=*/false);
    xp += 4;
    wp += (size_t)4 * M;
  }

  // D layout: VGPR v -> row (v + 8*half), col = lane&15 within the tile.
  float* op = Hout + (size_t)((tm << 4) + (half << 3)) * M + ((tn << 4) + l15);
  #pragma unroll
  for (int v = 0; v < 8; ++v) op[(size_t)v * M] = acc[v];
}

// ---------------------------------------------------------------------------
// a_s[n,h] = <h[n,h,:], att_src[h,:]>, a_d[n,h] = <h[n,h,:], att_dst[h,:]>
// One wave per (n,h); C=64 -> 2 elements/lane, shfl_xor tree reduction.
// ---------------------------------------------------------------------------
__global__ __launch_bounds__(256) void att_dots(
    const float* __restrict__ Hm, const float* __restrict__ att_s,
    const float* __restrict__ att_d, float* __restrict__ a_s,
    float* __restrict__ a_d, int Nrows, int Hh, int C)
{
  const int wave = blockIdx.x * (blockDim.x >> 5) + (threadIdx.x >> 5);
  if (wave >= Nrows * Hh) return;
  const int lane = threadIdx.x & 31;
  const int h = wave % Hh;
  const float* hv = Hm + (size_t)wave * C;
  const float* as = att_s + (size_t)h * C;
  const float* ad = att_d + (size_t)h * C;
  float ss = 0.f, sd = 0.f;
  for (int c = lane; c < C; c += 32) {
    float x = hv[c];
    ss += x * as[c];
    sd += x * ad[c];
  }
  #pragma unroll
  for (int off = 16; off; off >>= 1) {
    ss += __shfl_xor(ss, off, 32);
    sd += __shfl_xor(sd, off, 32);
  }
  if (lane == 0) { a_s[wave] = ss; a_d[wave] = sd; }
}

// ---------------------------------------------------------------------------
// Helpers / init
// ---------------------------------------------------------------------------
__device__ __forceinline__ float lrelu(float e) {
  return e > 0.f ? e : NEG_SLOPE * e;
}

__global__ void init_ms(float* __restrict__ m, float* __restrict__ s, int total)
{
  int i = blockIdx.x * blockDim.x + threadIdx.x;
  if (i < total) { m[i] = -__builtin_inff(); s[i] = 0.f; }
}

__global__ void init_bias(float* __restrict__ out, const float* __restrict__ bias,
                          int HC, int total)
{
  int i = blockIdx.x * blockDim.x + threadIdx.x;
  if (i < total) out[i] = bias[i % HC];
}

__global__ void elu_inplace(float* __restrict__ x, int total)
{
  int i = blockIdx.x * blockDim.x + threadIdx.x;
  if (i < total) { float v = x[i]; x[i] = v > 0.f ? v : expm1f(v); }
}

// ---------------------------------------------------------------------------
// Edge passes. Virtual edges [E, E+N) are the self-loops.
// ---------------------------------------------------------------------------
template <int H>
__global__ void edge_max(const int* __restrict__ src, const int* __restrict__ dst,
                         const float* __restrict__ a_s, const float* __restrict__ a_d,
                         float* __restrict__ m, int E, int N)
{
  int i = blockIdx.x * blockDim.x + threadIdx.x;
  if (i >= E + N) return;
  int si, di;
  if (i < E) { si = src[i]; di = dst[i]; } else { si = di = i - E; }
  #pragma unroll
  for (int h = 0; h < H; ++h) {
    float e = lrelu(a_s[si * H + h] + a_d[di * H + h]);
    __hip_atomic_fetch_max(&m[di * H + h], e,
                           __ATOMIC_RELAXED, __HIP_MEMORY_SCOPE_AGENT);
  }
}

template <int H>
__global__ void edge_expsum(const int* __restrict__ src, const int* __restrict__ dst,
                            const float* __restrict__ a_s, const float* __restrict__ a_d,
                            const float* __restrict__ m, float* __restrict__ ssum,
                            int E, int N)
{
  int i = blockIdx.x * blockDim.x + threadIdx.x;
  if (i >= E + N) return;
  int si, di;
  if (i < E) { si = src[i]; di = dst[i]; } else { si = di = i - E; }
  #pragma unroll
  for (int h = 0; h < H; ++h) {
    float e = lrelu(a_s[si * H + h] + a_d[di * H + h]);
    float w = __expf(e - m[di * H + h]);
    atomicAdd(&ssum[di * H + h], w);
  }
}

// out[dst, h, c] += h[src, h, c] * alpha(edge, h). 256 threads cover
// 256/(H*C) edges per block; each thread owns one (h,c) feature slot.
template <int H, int C>
__global__ __launch_bounds__(256) void edge_aggregate(
    const int* __restrict__ src, const int* __restrict__ dst,
    const float* __restrict__ a_s, const float* __restrict__ a_d,
    const float* __restrict__ m, const float* __restrict__ ssum,
    const float* __restrict__ Hfeat, float* __restrict__ out, int E, int N)
{
  constexpr int HC  = H * C;
  constexpr int EPB = 256 / HC;
  const int t    = threadIdx.x;
  const int eidx = blockIdx.x * EPB + t / HC;
  if (eidx >= E + N) return;
  const int f = t % HC;
  const int h = f / C;
  int si, di;
  if (eidx < E) { si = src[eidx]; di = dst[eidx]; } else { si = di = eidx - E; }
  float e = lrelu(a_s[si * H + h] + a_d[di * H + h]);
  float w = __expf(e - m[di * H + h]);
  float alpha = w / (ssum[di * H + h] + PYG_EPS);
  atomicAdd(&out[(size_t)di * HC + f], Hfeat[(size_t)si * HC + f] * alpha);
}

// ---------------------------------------------------------------------------
// Launcher
// ---------------------------------------------------------------------------
extern "C" void kernel_launch(void* const* d_in, const int* in_sizes, int n_in,
                              void* d_out, int out_size, void* d_ws, size_t ws_size,
                              hipStream_t stream)
{
  const float* x   = (const float*)d_in[0];
  const int*   ei  = (const int*)  d_in[1];
  const float* W1  = (const float*)d_in[2];
  const float* as1 = (const float*)d_in[3];
  const float* ad1 = (const float*)d_in[4];
  const float* b1  = (const float*)d_in[5];
  const float* W2  = (const float*)d_in[6];
  const float* as2 = (const float*)d_in[7];
  const float* ad2 = (const float*)d_in[8];
  const float* b2  = (const float*)d_in[9];
  float* out = (float*)d_out;

  const int F = 128, H1 = 4, C = 64, HC1 = 256, HC2 = 64;
  const int N  = in_sizes[0] / F;       // 50000
  const int E  = in_sizes[1] / 2;       // 800000
  const int EE = E + N;                 // + self loops
  const int* src = ei;                  // edge_index[0,:]
  const int* dst = ei + E;              // edge_index[1,:]

  // Workspace layout (fp32), base is 256B-aligned device memory.
  float* h1  = (float*)d_ws;                 // [N, 256]
  float* x2  = h1  + (size_t)N * HC1;        // [N, 256] layer-1 out -> ELU -> layer-2 in
  float* h2  = x2  + (size_t)N * HC1;        // [N, 64]
  float* aS1 = h2  + (size_t)N * HC2;        // [N, 4]
  float* aD1 = aS1 + (size_t)N * H1;
  float* m1  = aD1 + (size_t)N * H1;
  float* s1  = m1  + (size_t)N * H1;
  float* aS2 = s1  + (size_t)N * H1;         // [N, 1]
  float* aD2 = aS2 + N;
  float* m2  = aD2 + N;
  float* s2  = m2  + N;

  const int TB = 256;
  auto cdiv = [](int a, int b) { return (a + b - 1) / b; };

  // ---------------- Layer 1 ----------------
  {
    int tiles = (N / 16) * (HC1 / 16);
    gemm_wmma_f32<<<cdiv(tiles, 8), TB, 0, stream>>>(x, W1, h1, N, F, HC1);
  }
  att_dots<<<cdiv(N * H1, 8), TB, 0, stream>>>(h1, as1, ad1, aS1, aD1, N, H1, C);
  init_ms<<<cdiv(N * H1, TB), TB, 0, stream>>>(m1, s1, N * H1);
  init_bias<<<cdiv(N * HC1, TB), TB, 0, stream>>>(x2, b1, HC1, N * HC1);
  edge_max<H1><<<cdiv(EE, TB), TB, 0, stream>>>(src, dst, aS1, aD1, m1, E, N);
  edge_expsum<H1><<<cdiv(EE, TB), TB, 0, stream>>>(src, dst, aS1, aD1, m1, s1, E, N);
  edge_aggregate<H1, C><<<EE, TB, 0, stream>>>(src, dst, aS1, aD1, m1, s1, h1, x2, E, N);
  elu_inplace<<<cdiv(N * HC1, TB), TB, 0, stream>>>(x2, N * HC1);

  // ---------------- Layer 2 ----------------
  {
    int tiles = (N / 16) * (HC2 / 16);
    gemm_wmma_f32<<<cdiv(tiles, 8), TB, 0, stream>>>(x2, W2, h2, N, HC1, HC2);
  }
  att_dots<<<cdiv(N, 8), TB, 0, stream>>>(h2, as2, ad2, aS2, aD2, N, 1, C);
  init_ms<<<cdiv(N, TB), TB, 0, stream>>>(m2, s2, N);
  init_bias<<<cdiv(N * HC2, TB), TB, 0, stream>>>(out, b2, HC2, N * HC2);
  edge_max<1><<<cdiv(EE, TB), TB, 0, stream>>>(src, dst, aS2, aD2, m2, E, N);
  edge_expsum<1><<<cdiv(EE, TB), TB, 0, stream>>>(src, dst, aS2, aD2, m2, s2, E, N);
  edge_aggregate<1, C><<<cdiv(EE, 4), TB, 0, stream>>>(src, dst, aS2, aD2, m2, s2, h2, out, E, N);
}